// MatrixGraphConvolution_45019847197213
// MI455X (gfx1250) — compile-verified
//
#include <hip/hip_runtime.h>
#include <hip/hip_bf16.h>

// ---------------------------------------------------------------------------
// out = D^-1 A (x W^T) + x B^T ; A[dst,src]=1 (set semantics, duplicates
// collapse), deg[dst] counts every edge (index_add semantics).
//
// Deterministic pipeline:
//   0) zero count + N*N adjacency bitmap
//   1) WMMA f32 GEMM (LDS-staged A): xW = x@W^T, xB = x@B^T
//   2) per-edge: int atomicAdd(count[dst]) + atomicOr(bitmap[dst][src])
//      (both order-independent -> bit-deterministic)
//   3) per-dst wave: walk bitmap row in increasing src order, accumulate
//      xW rows in registers, fused epilogue out = acc/deg + xB
// ---------------------------------------------------------------------------

typedef __attribute__((ext_vector_type(2))) float v2f;
typedef __attribute__((ext_vector_type(8))) float v8f;

#define FDIM 128       // F_IN == F_OUT
#define XS_STRIDE 132  // padded LDS row stride (floats): 16B aligned, no bank conflicts

// ---------------- Phase 0: zero scratch ------------------------------------
__global__ __launch_bounds__(256) void zero_words(unsigned int* __restrict__ p,
                                                  long long nwords) {
  long long i = (long long)blockIdx.x * blockDim.x + threadIdx.x;
  const long long stride = (long long)gridDim.x * blockDim.x;
  for (; i < nwords; i += stride) p[i] = 0u;
}

// ---------------- Phase 1: dual GEMM via f32 WMMA --------------------------
// Block = 256 threads = 8 waves. Block owns 16 rows of x (staged in LDS once,
// shared by all 8 waves); wave w computes the 16x16 tile at column tile w for
// BOTH xW and xB, sharing the A fragment between the two WMMA chains.
// A (16x4 f32): lane m=lane&15 -> row m; lanes 0-15 K=0,1 ; lanes 16-31 K=2,3.
// B (4x16 f32): lane n=lane&15 -> col n; B[k][n]=W[n][k] -> contiguous float2.
// D (16x16 f32): VGPR v -> M = v + 8*(lane>=16), N = lane&15.
__global__ __launch_bounds__(256) void gemm_xWB(const float* __restrict__ x,
                                                const float* __restrict__ W,
                                                const float* __restrict__ Bmat,
                                                float* __restrict__ xW,
                                                float* __restrict__ xB) {
  __shared__ float xs[16 * XS_STRIDE];  // 8.25 KB

  const int lane = threadIdx.x & 31;
  const int wave = threadIdx.x >> 5;    // == colTile
  const int rowTile = blockIdx.x;

  // Stage the block's 16 x-rows into LDS (2 rows per wave, float4 per lane).
  {
    const float* srcX = x + (size_t)rowTile * 16 * FDIM;
    for (int r = wave; r < 16; r += 8) {
      const float4 v = *(const float4*)(srcX + r * FDIM + lane * 4);
      *(float4*)&xs[r * XS_STRIDE + lane * 4] = v;
    }
  }
  __syncthreads();

  const int half = lane >> 4;  // 0: K=0,1   1: K=2,3
  const int mn   = lane & 15;

  const float* aRow = &xs[mn * XS_STRIDE];
  const float* wRow = W    + (size_t)(wave * 16 + mn) * FDIM;
  const float* bRow = Bmat + (size_t)(wave * 16 + mn) * FDIM;

  v8f cW = {};
  v8f cB = {};

#pragma unroll 8
  for (int k = 0; k < FDIM; k += 4) {
    const int kk = k + 2 * half;
    v2f a  = *(const v2f*)(aRow + kk);   // ds_load_b64, bank-conflict-free
    v2f bw = *(const v2f*)(wRow + kk);
    v2f bb = *(const v2f*)(bRow + kk);
    cW = __builtin_amdgcn_wmma_f32_16x16x4_f32(false, a, false, bw,
                                               (short)0, cW, false, false);
    cB = __builtin_amdgcn_wmma_f32_16x16x4_f32(false, a, false, bb,
                                               (short)0, cB, false, false);
  }

  const int n = wave * 16 + mn;
#pragma unroll
  for (int v = 0; v < 8; ++v) {
    const int m = rowTile * 16 + v + 8 * half;
    xW[(size_t)m * FDIM + n] = cW[v];
    xB[(size_t)m * FDIM + n] = cB[v];
  }
}

// ---------------- Phase 2: edge pass (integer atomics only) ----------------
__global__ __launch_bounds__(256) void edge_pass(
    const long long* __restrict__ ei,  // [2,E] int64: row0=src, row1=dst
    int* __restrict__ count,
    unsigned int* __restrict__ bitmap,
    long long E, int N) {
  const long long i = (long long)blockIdx.x * blockDim.x + threadIdx.x;
  if (i >= E) return;
  const int src = (int)ei[i];
  const int dst = (int)ei[E + i];
  atomicAdd(&count[dst], 1);                       // deg: duplicates accumulate
  const size_t word = (size_t)dst * (size_t)(N >> 5) + (size_t)(src >> 5);
  atomicOr(&bitmap[word], 1u << (src & 31));       // A: set semantics
}

// ---------------- Phase 3: deterministic gather + fused epilogue -----------
// One wave per dst node. Walk the dst's bitmap row (uniform uint4 loads);
// for each set bit (increasing src order) accumulate xW[src] across lanes.
__global__ __launch_bounds__(256) void gather_nodes(
    const unsigned int* __restrict__ bitmap,
    const int* __restrict__ count,
    const float* __restrict__ xW,
    const float* __restrict__ xB,
    float* __restrict__ out,
    int N) {
  const int lane = threadIdx.x & 31;
  const int wave = threadIdx.x >> 5;
  const int d = blockIdx.x * 8 + wave;
  if (d >= N) return;

  const int rowWords = N >> 5;  // 512
  const uint4* bitRow = (const uint4*)(bitmap + (size_t)d * rowWords);

  float4 acc = make_float4(0.f, 0.f, 0.f, 0.f);
  const int q4 = rowWords >> 2;  // 128 uint4 per row
  for (int q = 0; q < q4; ++q) {
    const uint4 wv = bitRow[q];  // uniform across the wave
    unsigned wbits[4] = {wv.x, wv.y, wv.z, wv.w};
#pragma unroll
    for (int c = 0; c < 4; ++c) {
      unsigned wb = wbits[c];
      const int base = (q * 4 + c) << 5;
      while (wb) {
        const int b = __builtin_ctz(wb);
        wb &= wb - 1;
        const int s = base + b;  // srcs visited in increasing order
        const float4 v = *(const float4*)(xW + (size_t)s * FDIM + lane * 4);
        acc.x += v.x; acc.y += v.y; acc.z += v.z; acc.w += v.w;
      }
    }
  }

  const int cnt = count[d];
  const float inv = (cnt == 0) ? 1.0f : (1.0f / (float)cnt);
  const size_t o = (size_t)d * FDIM + lane * 4;
  const float4 b = *(const float4*)(xB + o);
  float4 r;
  r.x = acc.x * inv + b.x;
  r.y = acc.y * inv + b.y;
  r.z = acc.z * inv + b.z;
  r.w = acc.w * inv + b.w;
  *(float4*)(out + o) = r;
}

// ---------------------------------------------------------------------------
extern "C" void kernel_launch(void* const* d_in, const int* in_sizes, int n_in,
                              void* d_out, int out_size, void* d_ws, size_t ws_size,
                              hipStream_t stream) {
  const float* x      = (const float*)d_in[0];
  const long long* ei = (const long long*)d_in[1];
  const float* W      = (const float*)d_in[2];
  const float* Bmat   = (const float*)d_in[3];
  float* out          = (float*)d_out;

  const long long N64 = (long long)in_sizes[0] / FDIM;  // 16384
  const long long E   = (long long)in_sizes[1] / 2;     // 524288
  const int N = (int)N64;

  // Workspace layout (bytes):
  //   xW     : N*128*4  = 8 MB       @ 0
  //   xB     : N*128*4  = 8 MB       @ featBytes
  //   count  : N*4      = 64 KB      @ 2*featBytes      (zeroed, contiguous
  //   bitmap : N*N/8    = 32 MB      @ 2*featBytes+N*4   with bitmap)
  char* ws = (char*)d_ws;
  const size_t featBytes = (size_t)N * FDIM * sizeof(float);
  float* xW          = (float*)(ws);
  float* xB          = (float*)(ws + featBytes);
  int* count         = (int*)(ws + 2 * featBytes);
  unsigned int* bmap = (unsigned int*)(ws + 2 * featBytes + (size_t)N * sizeof(int));

  // Phase 0: zero count + bitmap (one contiguous region).
  const long long zeroWords =
      (long long)(((size_t)N * sizeof(int) + ((size_t)N * (size_t)N) / 8) /
                  sizeof(unsigned int));
  zero_words<<<4096, 256, 0, stream>>>((unsigned int*)count, zeroWords);

  // Phase 1: WMMA GEMM, one block per 16-row tile (8 waves = 8 column tiles).
  gemm_xWB<<<N / 16, 256, 0, stream>>>(x, W, Bmat, xW, xB);

  // Phase 2: one thread per edge, integer atomics only.
  edge_pass<<<(int)((E + 255) / 256), 256, 0, stream>>>(ei, count, bmap, E, N);

  // Phase 3: one wave per dst node, deterministic bitmap-ordered gather.
  gather_nodes<<<N / 8, 256, 0, stream>>>(bmap, count, xW, xB, out, N);
}